// FM_73907797229838
// MI455X (gfx1250) — compile-verified
//
#include <hip/hip_runtime.h>
#include <hip/hip_bf16.h>

#define FM_BATCH 2048
#define FM_DIM   100000
#define FM_K     16

// Split-K: 128 M-tiles x 25 K-blocks x 8 waves = 25600 waves.
// 100000 = (25*8) chunks * 125 quad-steps * 4 -- exact, no tail chunks.
#define FM_KSPLIT    25
#define FM_KQ_WAVE   125   // K quad-steps (K advances 4 per wmma pair) per wave

typedef __attribute__((ext_vector_type(2))) float v2f;
typedef __attribute__((ext_vector_type(8))) float v8f;

// ---------------------------------------------------------------------------
// Phase 0: zero the split-K accumulation workspace (harness poisons d_ws).
// ---------------------------------------------------------------------------
__global__ void fm_zero_ws(float* __restrict__ p, int n) {
    int i = blockIdx.x * blockDim.x + threadIdx.x;
    if (i < n) p[i] = 0.0f;
}

// ---------------------------------------------------------------------------
// Phase 1: split-K WMMA accumulation.
//   S1[b,k] += X @ v        (v_wmma_f32_16x16x4_f32)
//   S2[b,k] += X^2 @ v^2    (second wmma, operands squared in-register)
//   L[b]    += X @ w        (per-lane VALU FMA)
// One wave = one 16-row batch tile x one 500-element K chunk.
// nq is a RUNTIME arg so the compiler cannot fully unroll the K loop
// (round-1 disasm showed a 125x unroll blowing past 256 VGPRs).
// ---------------------------------------------------------------------------
__global__ void __launch_bounds__(256)
fm_phase1(const float* __restrict__ X,
          const float* __restrict__ w,
          const float* __restrict__ v,
          float* __restrict__ S1,
          float* __restrict__ S2,
          float* __restrict__ L,
          int nq) {
    const int lane  = threadIdx.x & 31;
    const int wave  = threadIdx.x >> 5;      // 0..7
    const int tileM = blockIdx.x;            // 0..127
    const int kblk  = blockIdx.y;            // 0..KSPLIT-1

    const int chunk = kblk * 8 + wave;       // 0..199
    const int k0    = chunk * (FM_KQ_WAVE * 4);

    const int half = lane >> 4;              // 0: K={k,k+1}, 1: K={k+2,k+3}
    const int l16  = lane & 15;

    // A-matrix (16x4 f32): lane (half,l16) holds X[tileM*16+l16][k0+2*half + {0,1}]
    const float* pA = X + (size_t)(tileM * 16 + l16) * FM_DIM + k0 + 2 * half;
    // B-matrix (4x16 f32): VGPR0 = v[k0+2*half+0][l16], VGPR1 = v[k0+2*half+1][l16]
    const float* pB = v + (size_t)(k0 + 2 * half) * FM_K + l16;
    // w pairs for the linear term (same K slots as A)
    const float* pw = w + k0 + 2 * half;

    v8f c1 = {};     // accumulates X @ v tile
    v8f c2 = {};     // accumulates X^2 @ v^2 tile
    float accw = 0.0f;

    // One K quad-step: 2 b64 + 2 b32 loads, 6 VALU, 2 WMMA.
    auto quad = [&]() {
        // Streamed X: non-temporal (819 MB, single use; keep v/w hot in L2)
        v2f a = __builtin_nontemporal_load((const v2f*)pA);
        float b0 = pB[0];
        float b1 = pB[FM_K];
        v2f wv;
        wv.x = pw[0];
        wv.y = pw[1];

        v2f a2 = a * a;                      // x^2 in-register: no 2nd read of X
        v2f b;  b.x = b0;       b.y = b1;
        v2f b2; b2.x = b0 * b0; b2.y = b1 * b1;

        accw += a.x * wv.x + a.y * wv.y;     // linear term partial

        // D = A*B + C   (8 args: neg_a, A, neg_b, B, c_mod, C, reuse_a, reuse_b)
        c1 = __builtin_amdgcn_wmma_f32_16x16x4_f32(
                 false, a,  false, b,  (short)0, c1, false, false);
        c2 = __builtin_amdgcn_wmma_f32_16x16x4_f32(
                 false, a2, false, b2, (short)0, c2, false, false);

        pA += 4;                             // K advances by 4
        pB += 4 * FM_K;
        pw += 4;
    };

    int it = 0;
    const int nq8 = nq & ~7;
    for (; it < nq8; it += 8) {
        // One prefetch per 8 quads (128 B of per-lane stream), 1 KB ahead,
        // high locality -> near-scope prefetch instead of SCOPE_SYS.
        __builtin_prefetch((const void*)(pA + 256), 0, 3);
#pragma unroll
        for (int u = 0; u < 8; ++u) quad();
    }
    for (; it < nq; ++it) quad();            // runtime tail (uniform: no EXEC divergence)

    // Epilogue: C/D layout -> VGPR r, lanes 0-15: (M=r, N=l16); lanes 16-31: (M=r+8, N=l16)
    const int mBase = tileM * 16 + half * 8;
#pragma unroll
    for (int r = 0; r < 8; ++r) {
        const int idx = (mBase + r) * FM_K + l16;
        atomicAdd(&S1[idx], c1[r]);          // global_atomic_add_f32
        atomicAdd(&S2[idx], c2[r]);
    }
    // Both half-groups hold partials for row l16 of this tile; atomics merge them.
    atomicAdd(&L[tileM * 16 + l16], accw);
}

// ---------------------------------------------------------------------------
// Phase 2: out[b] = L[b] + 0.5 * sum_k( S1[b,k]^2 - S2[b,k] )
// ---------------------------------------------------------------------------
__global__ void fm_phase2(const float* __restrict__ S1,
                          const float* __restrict__ S2,
                          const float* __restrict__ L,
                          float* __restrict__ out) {
    int b = blockIdx.x * blockDim.x + threadIdx.x;
    if (b < FM_BATCH) {
        float s = 0.0f;
#pragma unroll
        for (int k = 0; k < FM_K; ++k) {
            float x = S1[b * FM_K + k];
            s += x * x - S2[b * FM_K + k];
        }
        out[b] = L[b] + 0.5f * s;
    }
}

extern "C" void kernel_launch(void* const* d_in, const int* in_sizes, int n_in,
                              void* d_out, int out_size, void* d_ws, size_t ws_size,
                              hipStream_t stream) {
    const float* X = (const float*)d_in[0];   // [2048, 100000]
    const float* w = (const float*)d_in[1];   // [100000, 1]
    const float* v = (const float*)d_in[2];   // [100000, 16]
    float* out = (float*)d_out;               // [2048]

    float* S1 = (float*)d_ws;                 // [2048, 16]
    float* S2 = S1 + FM_BATCH * FM_K;         // [2048, 16]
    float* L  = S2 + FM_BATCH * FM_K;         // [2048]

    const int nzero = FM_BATCH * FM_K * 2 + FM_BATCH;
    fm_zero_ws<<<(nzero + 255) / 256, 256, 0, stream>>>(S1, nzero);

    dim3 grid(FM_BATCH / 16, FM_KSPLIT);      // 128 x 25 blocks, 8 waves each
    fm_phase1<<<grid, 256, 0, stream>>>(X, w, v, S1, S2, L, FM_KQ_WAVE);

    fm_phase2<<<(FM_BATCH + 255) / 256, 256, 0, stream>>>(S1, S2, L, out);
}